// GridFeaturesGroupIntraCommunication_45002667327774
// MI455X (gfx1250) — compile-verified
//
#include <hip/hip_runtime.h>
#include <cstdint>

#define PN 524288   // D*H*W for every volume (128*128*32)
#define CN 32       // channels == wave32 lanes

// ---------------------------------------------------------------------------
// float atomic min/max (exact, order independent) via CAS
// ---------------------------------------------------------------------------
__device__ inline void atomicMinF(float* addr, float val) {
  int* a = (int*)addr;
  int old = *a;
  while (__int_as_float(old) > val) {
    int prev = atomicCAS(a, old, __float_as_int(val));
    if (prev == old) break;
    old = prev;
  }
}
__device__ inline void atomicMaxF(float* addr, float val) {
  int* a = (int*)addr;
  int old = *a;
  while (__int_as_float(old) < val) {
    int prev = atomicCAS(a, old, __float_as_int(val));
    if (prev == old) break;
    old = prev;
  }
}

// ---------------------------------------------------------------------------
// ws minmax slots: mm[g*8 + 0..2] = min(xyz), mm[g*8 + 4..6] = max(xyz)
// ---------------------------------------------------------------------------
__global__ void init_minmax_kernel(float* mm) {
  int t = threadIdx.x;
  if (t < 24) {
    int g = t >> 3, k = t & 7;
    mm[g * 8 + k] = (k < 4) ? 3.402823466e38f : -3.402823466e38f;
  }
}

__global__ void minmax_kernel(const float* __restrict__ verts, float* __restrict__ mm) {
  __shared__ float red[256 * 8];
  float mn0 = 3.4e38f, mn1 = 3.4e38f, mn2 = 3.4e38f;
  float mx0 = -3.4e38f, mx1 = -3.4e38f, mx2 = -3.4e38f;
  for (long p = (long)blockIdx.x * blockDim.x + threadIdx.x; p < PN;
       p += (long)gridDim.x * blockDim.x) {
    float v0 = verts[p * 3 + 0], v1 = verts[p * 3 + 1], v2 = verts[p * 3 + 2];
    mn0 = fminf(mn0, v0); mx0 = fmaxf(mx0, v0);
    mn1 = fminf(mn1, v1); mx1 = fmaxf(mx1, v1);
    mn2 = fminf(mn2, v2); mx2 = fmaxf(mx2, v2);
  }
  int t = threadIdx.x;
  red[t * 8 + 0] = mn0; red[t * 8 + 1] = mn1; red[t * 8 + 2] = mn2;
  red[t * 8 + 4] = mx0; red[t * 8 + 5] = mx1; red[t * 8 + 6] = mx2;
  __syncthreads();
  for (int off = 128; off > 0; off >>= 1) {
    if (t < off) {
      for (int k = 0; k < 3; ++k) {
        red[t * 8 + k]     = fminf(red[t * 8 + k],     red[(t + off) * 8 + k]);
        red[t * 8 + 4 + k] = fmaxf(red[t * 8 + 4 + k], red[(t + off) * 8 + 4 + k]);
      }
    }
    __syncthreads();
  }
  if (t == 0) {
    for (int k = 0; k < 3; ++k) {
      atomicMinF(&mm[k], red[k]);
      atomicMaxF(&mm[4 + k], red[4 + k]);
    }
  }
}

// ---------------------------------------------------------------------------
// Transpose [C, S] -> [S, C] via Tensor Data Mover (LDS pad => 65-DW pitch)
// ---------------------------------------------------------------------------
#if defined(__gfx1250__) && __has_builtin(__builtin_amdgcn_tensor_load_to_lds) && \
    __has_builtin(__builtin_amdgcn_s_wait_tensorcnt)
#define USE_TDM 1
typedef unsigned int u32x4 __attribute__((ext_vector_type(4)));
typedef int i32x8 __attribute__((ext_vector_type(8)));
typedef int i32x4 __attribute__((ext_vector_type(4)));
#else
#define USE_TDM 0
#endif

__global__ void transpose_kernel(const float* __restrict__ src, float* __restrict__ dst) {
  __shared__ float tile[32 * 65];  // 32 channel rows, 65-DWORD pitch (TDM pad)
  const int s0 = blockIdx.x * 64;  // 64 spatial elements per tile

#if USE_TDM
  if (threadIdx.x < 32) {  // one TDM op per workgroup, issued by wave 0
    unsigned long long ga = (unsigned long long)(uintptr_t)(src + s0);
    unsigned int lds_off = (unsigned int)(uintptr_t)&tile[0];  // low 32b = LDS byte offset

    u32x4 g0;
    g0[0] = 1u;                                           // count=1 (valid user D#)
    g0[1] = lds_off;                                      // lds_addr
    g0[2] = (unsigned int)(ga & 0xFFFFFFFFu);             // global_addr[31:0]
    g0[3] = (unsigned int)((ga >> 32) & 0x1FFFFFFu)       // global_addr[56:32]
            | (2u << 30);                                 // type = 2 (image/tensor)

    i32x8 g1;
    g1[0] = (2 << 16)      // data_size = 4 bytes
          | (1 << 20)      // pad_enable
          | (5 << 22)      // pad_interval: every 64 DWORDs
          | (0 << 25);     // pad_amount: 1 DWORD  -> 65-DW row pitch
    g1[1] = (64 << 16);    // tensor_dim0[15:0] = 64 (tile fully in-bounds)
    g1[2] = (32 << 16);    // tensor_dim0 hi = 0 | tensor_dim1[15:0] = 32
    g1[3] = (64 << 16);    // tensor_dim1 hi = 0 | tile_dim0 = 64 (spatial run)
    g1[4] = 32;            // tile_dim1 = 32 channels | tile_dim2 = 0 (2-D)
    g1[5] = PN;            // tensor_dim0_stride = 524288 elems (channel stride)
    g1[6] = 0;
    g1[7] = 0;

    i32x4 z4; z4[0] = 0; z4[1] = 0; z4[2] = 0; z4[3] = 0;   // groups 2/3 unused (2-D)
    i32x8 z8; z8[0] = 0; z8[1] = 0; z8[2] = 0; z8[3] = 0;
    z8[4] = 0; z8[5] = 0; z8[6] = 0; z8[7] = 0;             // 6-arg form: extra group
    __builtin_amdgcn_tensor_load_to_lds(g0, g1, z4, z4, z8, 0);
    __builtin_amdgcn_s_wait_tensorcnt(0);
  }
  __syncthreads();
#else
  // Fallback: coalesced 256B row loads into padded LDS tile (read-once -> NT)
  for (int k = 0; k < 8; ++k) {
    int c = 4 * k + (threadIdx.x >> 6);
    int x = threadIdx.x & 63;
    tile[c * 65 + x] = __builtin_nontemporal_load(&src[(long)c * PN + s0 + x]);
  }
  __syncthreads();
#endif

  // Write phase: dst[(s0+s)*32 + c], contiguous in c then s -> 128B/wave stores.
  // Regular (RT) stores: featT is the gather working set we WANT resident in L2.
  // LDS read tile[c*65 + s]: lane varies c, stride 65 -> conflict-free.
  for (int k = 0; k < 8; ++k) {
    int idx = (int)threadIdx.x + k * 256;  // 0..2047
    int s = idx >> 5;
    int c = idx & 31;
    dst[(long)(s0 + s) * CN + c] = tile[c * 65 + s];
  }
}

// ---------------------------------------------------------------------------
// Trilinear sample of one channel (lane); CS/SS are COMPILE-TIME channel /
// spatial element strides (transposed layout: CS=1, SS=32 => addr = idx<<5+lane,
// pure 32-bit shift math).
// ---------------------------------------------------------------------------
template <int CS, int SS>
__device__ inline float tri_sample(const float* __restrict__ f,
                                   float gx, float gy, float gz,
                                   int W, int H, int D, int lane) {
  float ix = (gx + 1.0f) * 0.5f * (float)(W - 1);
  float iy = (gy + 1.0f) * 0.5f * (float)(H - 1);
  float iz = (gz + 1.0f) * 0.5f * (float)(D - 1);
  float x0f = floorf(ix), y0f = floorf(iy), z0f = floorf(iz);
  float fx = ix - x0f, fy = iy - y0f, fz = iz - z0f;
  int x0 = (int)x0f, y0 = (int)y0f, z0 = (int)z0f;
  x0 = x0 < 0 ? 0 : (x0 > W - 1 ? W - 1 : x0);
  y0 = y0 < 0 ? 0 : (y0 > H - 1 ? H - 1 : y0);
  z0 = z0 < 0 ? 0 : (z0 > D - 1 ? D - 1 : z0);
  int x1 = x0 + 1 > W - 1 ? W - 1 : x0 + 1;
  int y1 = y0 + 1 > H - 1 ? H - 1 : y0 + 1;
  int z1 = z0 + 1 > D - 1 ? D - 1 : z0 + 1;

  const long lc = (long)lane * CS;
  const long r00 = (long)((z0 * H + y0) * W);
  const long r10 = (long)((z0 * H + y1) * W);
  const long r01 = (long)((z1 * H + y0) * W);
  const long r11 = (long)((z1 * H + y1) * W);
  // Each load: 32 lanes * 4B contiguous = one 128B segment (transposed layout)
  float v000 = f[(r00 + x0) * SS + lc];
  float v001 = f[(r00 + x1) * SS + lc];
  float v010 = f[(r10 + x0) * SS + lc];
  float v011 = f[(r10 + x1) * SS + lc];
  float v100 = f[(r01 + x0) * SS + lc];
  float v101 = f[(r01 + x1) * SS + lc];
  float v110 = f[(r11 + x0) * SS + lc];
  float v111 = f[(r11 + x1) * SS + lc];

  float gx0 = 1.0f - fx, gy0 = 1.0f - fy, gz0 = 1.0f - fz;
  return v000 * (gz0 * gy0 * gx0) + v001 * (gz0 * gy0 * fx)
       + v010 * (gz0 * fy  * gx0) + v011 * (gz0 * fy  * fx)
       + v100 * (fz  * gy0 * gx0) + v101 * (fz  * gy0 * fx)
       + v110 * (fz  * fy  * gx0) + v111 * (fz  * fy  * fx);
}

// ---------------------------------------------------------------------------
// Main gather kernel: block = 8 waves * 32 lanes; wave handles 8 points,
// lane = channel. Results staged in LDS (pitch 33) then written out as
// contiguous [C][P] rows with NON-TEMPORAL stores (write-once output must not
// evict the 192MB L2-resident transposed feature volumes).
// ---------------------------------------------------------------------------
template <int CS, int SS>
__global__ void __launch_bounds__(256)
gather_kernel(const float* __restrict__ verts, const float* __restrict__ mm,
              const float* __restrict__ selfF,
              const float* __restrict__ srcA, const float* __restrict__ srcB,
              int WA, int HA, int DA, int WB, int HB, int DB,
              float* __restrict__ out) {
  __shared__ float tile[64 * 33];
  const int lane = threadIdx.x & 31;
  const int wave = threadIdx.x >> 5;
  const int p0 = blockIdx.x * 64;

  const float mn0 = mm[0], mn1 = mm[1], mn2 = mm[2];
  const float r0 = 2.0f / (mm[4] - mn0);
  const float r1 = 2.0f / (mm[5] - mn1);
  const float r2 = 2.0f / (mm[6] - mn2);

  for (int r = 0; r < 8; ++r) {
    const int pl = wave + r * 8;
    const long p = p0 + pl;
    const float gx = (verts[p * 3 + 0] - mn0) * r0 - 1.0f;
    const float gy = (verts[p * 3 + 1] - mn1) * r1 - 1.0f;
    const float gz = (verts[p * 3 + 2] - mn2) * r2 - 1.0f;

    float acc = selfF[p * SS + (long)lane * CS];
    acc += tri_sample<CS, SS>(srcA, gx, gy, gz, WA, HA, DA, lane);
    acc += tri_sample<CS, SS>(srcB, gx, gy, gz, WB, HB, DB, lane);
    tile[pl * 33 + lane] = acc;
  }
  __syncthreads();

  // out[(c)*P + p0 + s]; wave-contiguous 128B NT stores; LDS stride-33 reads.
  for (int k = 0; k < 8; ++k) {
    int c = 4 * k + ((int)threadIdx.x >> 6);
    int s = (int)threadIdx.x & 63;
    __builtin_nontemporal_store(tile[s * 33 + c], &out[(long)c * PN + p0 + s]);
  }
}

// ---------------------------------------------------------------------------
// Host launcher
// ---------------------------------------------------------------------------
extern "C" void kernel_launch(void* const* d_in, const int* in_sizes, int n_in,
                              void* d_out, int out_size, void* d_ws, size_t ws_size,
                              hipStream_t stream) {
  (void)in_sizes; (void)n_in; (void)out_size;
  const float* feat[3]  = {(const float*)d_in[0], (const float*)d_in[1], (const float*)d_in[2]};
  const float* verts[3] = {(const float*)d_in[3], (const float*)d_in[4], (const float*)d_in[5]};
  float* out = (float*)d_out;

  float* mm = (float*)d_ws;
  const size_t volBytes = (size_t)CN * PN * sizeof(float);   // 64 MB per volume
  const size_t need = 256 + 3 * volBytes;                    // ~192 MB transposed copies
  const bool useT = ws_size >= need;
  float* fT[3] = {
    (float*)((char*)d_ws + 256),
    (float*)((char*)d_ws + 256 + volBytes),
    (float*)((char*)d_ws + 256 + 2 * volBytes),
  };

  // 1) min/max normalization bounds (exact, deterministic)
  hipLaunchKernelGGL(init_minmax_kernel, dim3(1), dim3(32), 0, stream, mm);
  for (int i = 0; i < 3; ++i)
    hipLaunchKernelGGL(minmax_kernel, dim3(256), dim3(256), 0, stream, verts[i], mm + i * 8);

  // 2) transpose [C,S] -> [S,C] so corner channels are one 128B segment
  if (useT)
    for (int i = 0; i < 3; ++i)
      hipLaunchKernelGGL(transpose_kernel, dim3(PN / 64), dim3(256), 0, stream, feat[i], fT[i]);

  // 3) gather + sum, one launch per grid i
  const int Wd[3] = {32, 128, 128};
  const int Hd[3] = {128, 32, 128};
  const int Dd[3] = {128, 128, 32};
  for (int i = 0; i < 3; ++i) {
    const int j1 = (i == 0) ? 1 : 0;
    const int j2 = (i == 2) ? 1 : 2;
    float* outi = out + (size_t)i * CN * PN;
    if (useT) {
      hipLaunchKernelGGL((gather_kernel<1, CN>), dim3(PN / 64), dim3(256), 0, stream,
                         verts[i], mm + i * 8, fT[i], fT[j1], fT[j2],
                         Wd[j1], Hd[j1], Dd[j1], Wd[j2], Hd[j2], Dd[j2], outi);
    } else {
      hipLaunchKernelGGL((gather_kernel<PN, 1>), dim3(PN / 64), dim3(256), 0, stream,
                         verts[i], mm + i * 8, feat[i], feat[j1], feat[j2],
                         Wd[j1], Hd[j1], Dd[j1], Wd[j2], Hd[j2], Dd[j2], outi);
    }
  }
}